// RelationalLatentDynamics_82849919140105
// MI455X (gfx1250) — compile-verified
//
#include <hip/hip_runtime.h>

typedef __attribute__((ext_vector_type(2))) float v2f;
typedef __attribute__((ext_vector_type(8))) float v8f;

#define O_N   32
#define NPAIR 496         // 32*31/2
#define BATCH 256
#define TEF   512         // T*E
#define BO    8192        // BATCH*O_N

// ---------------------------------------------------------------------------
// Register-blocked fp32 WMMA: MT x NT tiles of 16x16, K in steps of 4.
// A row-major (lda), W row-major [K, N] (ldw). K multiple of 4.
// A-frag: lane m=lane&15, VGPR pair holds K = (lane>=16 ? {2,3} : {0,1}).
// B-frag mirrors with lane = column. Per K-step: MT A-frags + NT B-frags
// feed MT*NT v_wmma_f32_16x16x4_f32 (A reused NT times, B reused MT times).
// ---------------------------------------------------------------------------
template <int MT, int NT>
__device__ __forceinline__ void wmma_tiles(const float* A, int lda,
                                           const float* W, int ldw,
                                           int row0, int col0, int K,
                                           v8f acc[MT][NT]) {
  const int lane = threadIdx.x & 31;
  const int mn   = lane & 15;
  const int koff = (lane >> 4) << 1;                 // 0 or 2
  const float* ap = A + (size_t)(row0 + mn) * lda + koff;
  const float* bp = W + (size_t)koff * ldw + col0 + mn;
  for (int k = 0; k < K; k += 4) {
    v2f a[MT], b[NT];
#pragma unroll
    for (int mt = 0; mt < MT; ++mt) {
      a[mt].x = ap[(size_t)mt * 16 * lda + k];
      a[mt].y = ap[(size_t)mt * 16 * lda + k + 1];
    }
#pragma unroll
    for (int nt = 0; nt < NT; ++nt) {
      b[nt].x = bp[(size_t)k * ldw + nt * 16];
      b[nt].y = bp[(size_t)(k + 1) * ldw + nt * 16];
    }
#pragma unroll
    for (int mt = 0; mt < MT; ++mt)
#pragma unroll
      for (int nt = 0; nt < NT; ++nt)
        acc[mt][nt] = __builtin_amdgcn_wmma_f32_16x16x4_f32(
            false, a[mt], false, b[nt], (short)0, acc[mt][nt], false, false);
  }
}

// ---------------------------------------------------------------------------
// Generic GEMM: C[M,N] (+)= act(A[M,K] @ W[K,N] + bias)
// grid = (M/(16*MT), N/(16*NT)), block = 32 (one wave)
// flags: bit0 = relu, bit1 = accumulate into existing C
// ---------------------------------------------------------------------------
template <int MT, int NT>
__global__ void gemm_kernel(const float* __restrict__ A, int lda,
                            const float* __restrict__ W, int ldw,
                            const float* __restrict__ bias,
                            float* __restrict__ C, int ldc,
                            int K, int flags) {
  const int row0 = blockIdx.x * (16 * MT);
  const int col0 = blockIdx.y * (16 * NT);
  v8f acc[MT][NT] = {};
  wmma_tiles<MT, NT>(A, lda, W, ldw, row0, col0, K, acc);
  const int lane = threadIdx.x & 31;
  const int n    = lane & 15;
  const int mh   = (lane >> 4) * 8;
#pragma unroll
  for (int mt = 0; mt < MT; ++mt)
#pragma unroll
    for (int nt = 0; nt < NT; ++nt) {
      const int col  = col0 + nt * 16 + n;
      const float bv = bias ? bias[col] : 0.0f;
#pragma unroll
      for (int r = 0; r < 8; ++r) {
        float v = acc[mt][nt][r] + bv;
        float* cp = C + (size_t)(row0 + mt * 16 + mh + r) * ldc + col;
        if (flags & 2) v += *cp;
        if (flags & 1) v = fmaxf(v, 0.0f);
        *cp = v;
      }
    }
}

// ---------------------------------------------------------------------------
// Fused pair-edge kernel: per batch element b,
//   h1 = relu(projT[i] + projB[j] + b1)       (pair-decomposed first layer)
//   h2 = relu(h1 @ W2 + b2)     e = h2 @ W3 + b3
//   agg[i] += e ; agg[j] -= e   (atomic antisymmetric scatter)
// grid.x = BATCH, block = 256 (8 waves); each wave owns 16-pair chunks.
// W2/W3 staged in LDS once per block; h1/h2 tiles staged per wave.
// ---------------------------------------------------------------------------
__global__ void __launch_bounds__(256)
pair_edge_kernel(const float* __restrict__ projT, const float* __restrict__ projB,
                 const float* __restrict__ b1,
                 const float* __restrict__ W2, const float* __restrict__ b2,
                 const float* __restrict__ W3, const float* __restrict__ b3,
                 float* __restrict__ agg) {
  __shared__ int   s_pi[NPAIR], s_pj[NPAIR];
  __shared__ float s_W2[64 * 64];
  __shared__ float s_W3[64 * 64];
  __shared__ float s_h1[8][16 * 64];
  __shared__ float s_h2[8][16 * 64];

  // triu_indices(32, 1) table
  for (int p = threadIdx.x; p < NPAIR; p += blockDim.x) {
    int i = 0, rem = p;
    while (rem >= O_N - 1 - i) { rem -= O_N - 1 - i; ++i; }
    s_pi[p] = i;
    s_pj[p] = i + 1 + rem;
  }
  // stage weights (16 KB each) in LDS
  for (int t = threadIdx.x; t < 64 * 64; t += blockDim.x) {
    s_W2[t] = W2[t];
    s_W3[t] = W3[t];
  }
  __syncthreads();

  const int b    = blockIdx.x;
  const int wave = threadIdx.x >> 5;
  const int lane = threadIdx.x & 31;
  const int n    = lane & 15;
  const int mh   = (lane >> 4) * 8;
  float* h1 = s_h1[wave];
  float* h2 = s_h2[wave];
  const size_t base = (size_t)b * O_N * 64;

  for (int c = wave; c < NPAIR / 16; c += 8) {   // 31 chunks of 16 pairs
    const int p0 = c * 16;

    // stage h1 tile (16 pairs x 64) in LDS
    for (int t = lane; t < 16 * 64; t += 32) {
      const int r = t >> 6, col = t & 63;
      const int p = p0 + r;
      float v = projT[base + (size_t)s_pi[p] * 64 + col]
              + projB[base + (size_t)s_pj[p] * 64 + col] + b1[col];
      h1[t] = fmaxf(v, 0.0f);
    }
    asm volatile("s_wait_dscnt 0" ::: "memory");

    // h2 = relu(h1 @ W2 + b2)   (16x64 @ 64x64)
    {
      v8f acc[1][4] = {};
      wmma_tiles<1, 4>(h1, 64, s_W2, 64, 0, 0, 64, acc);
#pragma unroll
      for (int nt = 0; nt < 4; ++nt) {
        const int col = nt * 16 + n;
#pragma unroll
        for (int r = 0; r < 8; ++r)
          h2[(mh + r) * 64 + col] = fmaxf(acc[0][nt][r] + b2[col], 0.0f);
      }
    }
    asm volatile("s_wait_dscnt 0" ::: "memory");

    // e = h2 @ W3 + b3 ; antisymmetric atomic scatter
    {
      v8f acc[1][4] = {};
      wmma_tiles<1, 4>(h2, 64, s_W3, 64, 0, 0, 64, acc);
#pragma unroll
      for (int nt = 0; nt < 4; ++nt) {
        const int col = nt * 16 + n;
#pragma unroll
        for (int r = 0; r < 8; ++r) {
          const int p = p0 + mh + r;
          const float v = acc[0][nt][r] + b3[col];
          atomicAdd(agg + base + (size_t)s_pi[p] * 64 + col,  v);
          atomicAdd(agg + base + (size_t)s_pj[p] * 64 + col, -v);
        }
      }
    }
  }
}

// ---------------------------------------------------------------------------
// Elementwise kernels
// ---------------------------------------------------------------------------
__global__ void build_src_kernel(const float* __restrict__ z, float* __restrict__ src) {
  const int idx = blockIdx.x * blockDim.x + threadIdx.x;
  if (idx >= BO * TEF) return;
  const int f  = idx & (TEF - 1);
  const int bo = idx >> 9;
  const int o  = bo & 31;
  const int b  = bo >> 5;
  const int t  = f >> 5;
  const int e  = f & 31;
  const size_t zi = (((size_t)b * 16 + t) * O_N + o) * 32 + e;
  float v = z[zi];
  if (t > 0) v -= z[zi - (size_t)O_N * 32];     // temporal diff
  src[idx] = v;
}

__global__ void zero_kernel(float* __restrict__ p, int nel) {
  const int i = blockIdx.x * blockDim.x + threadIdx.x;
  if (i < nel) p[i] = 0.0f;
}

__global__ void build_zcur_kernel(const float* __restrict__ z,
                                  const float* __restrict__ zimpl,
                                  float* __restrict__ zcur) {
  const int idx = blockIdx.x * blockDim.x + threadIdx.x;
  if (idx >= BO * 64) return;
  const int c  = idx & 63;
  const int bo = idx >> 6;
  const int o  = bo & 31;
  const int b  = bo >> 5;
  float v;
  if (c < 32) v = z[(((size_t)b * 16 + 15) * O_N + o) * 32 + c];  // z[:, -1]
  else        v = zimpl[(size_t)bo * 32 + (c - 32)];
  zcur[idx] = v;
}

__global__ void update_kernel(float* __restrict__ zcur, const float* __restrict__ delta,
                              float* __restrict__ out, int t, int t_future) {
  const int idx = blockIdx.x * blockDim.x + threadIdx.x;
  if (idx >= BO * 64) return;
  const float v = zcur[idx] + delta[idx];
  zcur[idx] = v;
  const int c  = idx & 63;
  const int bo = idx >> 6;
  const int o  = bo & 31;
  const int b  = bo >> 5;
  if (c < 32)
    out[(((size_t)b * t_future + t) * O_N + o) * 32 + c] = v;
}

// ---------------------------------------------------------------------------
// Host orchestration
// ---------------------------------------------------------------------------
extern "C" void kernel_launch(void* const* d_in, const int* in_sizes, int n_in,
                              void* d_out, int out_size, void* d_ws, size_t ws_size,
                              hipStream_t stream) {
  (void)in_sizes; (void)n_in; (void)ws_size;
  const float* z     = (const float*)d_in[0];
  const float* ee1_w = (const float*)d_in[1];  const float* ee1_b = (const float*)d_in[2];
  const float* ee2_w = (const float*)d_in[3];  const float* ee2_b = (const float*)d_in[4];
  const float* ee3_w = (const float*)d_in[5];  const float* ee3_b = (const float*)d_in[6];
  const float* ne1_w = (const float*)d_in[7];  const float* ne1_b = (const float*)d_in[8];
  const float* ne2_w = (const float*)d_in[9];  const float* ne2_b = (const float*)d_in[10];
  const float* ne3_w = (const float*)d_in[11]; const float* ne3_b = (const float*)d_in[12];
  const float* ne4_w = (const float*)d_in[13]; const float* ne4_b = (const float*)d_in[14];
  const float* le1_w = (const float*)d_in[15]; const float* le1_b = (const float*)d_in[16];
  const float* le2_w = (const float*)d_in[17]; const float* le2_b = (const float*)d_in[18];
  const float* le3_w = (const float*)d_in[19]; const float* le3_b = (const float*)d_in[20];
  const float* lt1_w = (const float*)d_in[21]; const float* lt1_b = (const float*)d_in[22];
  const float* lt2_w = (const float*)d_in[23]; const float* lt2_b = (const float*)d_in[24];
  const float* lt3_w = (const float*)d_in[25]; const float* lt3_b = (const float*)d_in[26];
  const float* lt4_w = (const float*)d_in[27]; const float* lt4_b = (const float*)d_in[28];
  const float* lt5_w = (const float*)d_in[29]; const float* lt5_b = (const float*)d_in[30];
  float* out = (float*)d_out;
  const int t_future = out_size / (BATCH * O_N * 32);   // derive on host

  // workspace layout (floats)
  float* ws    = (float*)d_ws;
  float* src   = ws;                        // BO * 512
  float* projT = src   + (size_t)BO * TEF;  // BO * 64
  float* projB = projT + (size_t)BO * 64;
  float* agg   = projB + (size_t)BO * 64;
  float* h1    = agg   + (size_t)BO * 64;
  float* h2    = h1    + (size_t)BO * 64;
  float* h3    = h2    + (size_t)BO * 64;   // used with ld=32
  float* zimpl = h3    + (size_t)BO * 64;   // ld = 32
  float* zcur  = zimpl + (size_t)BO * 64;   // BO * 64
  float* delta = zcur  + (size_t)BO * 64;

  const dim3 wblk(32);
  const dim3 g2(BO / 32, 1);    // MT=2 kernels: 256 waves over M=8192
  const int  EW = 256;

  // 1) src = [z0, diffs] flattened
  build_src_kernel<<<(BO * TEF + EW - 1) / EW, EW, 0, stream>>>(z, src);

  // 2) ee pair-layer decomposition: projT/projB = src @ ee1_w halves
  gemm_kernel<2, 4><<<g2, wblk, 0, stream>>>(src, TEF, ee1_w,            64, nullptr, projT, 64, 512, 0);
  gemm_kernel<2, 4><<<g2, wblk, 0, stream>>>(src, TEF, ee1_w + 512 * 64, 64, nullptr, projB, 64, 512, 0);
  zero_kernel<<<(BO * 64 + EW - 1) / EW, EW, 0, stream>>>(agg, BO * 64);
  pair_edge_kernel<<<BATCH, 256, 0, stream>>>(projT, projB, ee1_b, ee2_w, ee2_b, ee3_w, ee3_b, agg);

  // 3) ne MLP: concat(src, agg) -> 64 -> 64 -> 32 -> 32
  gemm_kernel<2, 4><<<g2, wblk, 0, stream>>>(src, TEF, ne1_w,            64, nullptr, h1, 64, 512, 0);
  gemm_kernel<2, 4><<<g2, wblk, 0, stream>>>(agg, 64,  ne1_w + 512 * 64, 64, ne1_b,   h1, 64, 64, 3);
  gemm_kernel<2, 4><<<g2, wblk, 0, stream>>>(h1, 64, ne2_w, 64, ne2_b, h2,    64, 64, 1);
  gemm_kernel<2, 2><<<g2, wblk, 0, stream>>>(h2, 64, ne3_w, 32, ne3_b, h3,    32, 64, 1);
  gemm_kernel<2, 2><<<g2, wblk, 0, stream>>>(h3, 32, ne4_w, 32, ne4_b, zimpl, 32, 32, 0);

  // 4) z_cur = [z[:, -1], z_impl]
  build_zcur_kernel<<<(BO * 64 + EW - 1) / EW, EW, 0, stream>>>(z, zimpl, zcur);

  // 5) recurrent rollout
  for (int t = 0; t < t_future; ++t) {
    gemm_kernel<2, 4><<<g2, wblk, 0, stream>>>(zcur, 64, le1_w,           64, nullptr, projT, 64, 64, 0);
    gemm_kernel<2, 4><<<g2, wblk, 0, stream>>>(zcur, 64, le1_w + 64 * 64, 64, nullptr, projB, 64, 64, 0);
    zero_kernel<<<(BO * 64 + EW - 1) / EW, EW, 0, stream>>>(agg, BO * 64);
    pair_edge_kernel<<<BATCH, 256, 0, stream>>>(projT, projB, le1_b, le2_w, le2_b, le3_w, le3_b, agg);

    // lt MLP: concat(zcur, agg) -> 64 -> 64 -> 32 -> 16 -> 64
    gemm_kernel<2, 4><<<g2, wblk, 0, stream>>>(zcur, 64, lt1_w,           64, nullptr, h1, 64, 64, 0);
    gemm_kernel<2, 4><<<g2, wblk, 0, stream>>>(agg,  64, lt1_w + 64 * 64, 64, lt1_b,   h1, 64, 64, 3);
    gemm_kernel<2, 4><<<g2, wblk, 0, stream>>>(h1, 64, lt2_w, 64, lt2_b, h2, 64, 64, 1);
    gemm_kernel<2, 2><<<g2, wblk, 0, stream>>>(h2, 64, lt3_w, 32, lt3_b, h3, 32, 64, 1);
    gemm_kernel<2, 1><<<g2, wblk, 0, stream>>>(h3, 32, lt4_w, 16, lt4_b, h1, 16, 32, 1);   // h1 reused, ld=16
    gemm_kernel<2, 4><<<g2, wblk, 0, stream>>>(h1, 16, lt5_w, 64, lt5_b, delta, 64, 16, 0);

    update_kernel<<<(BO * 64 + EW - 1) / EW, EW, 0, stream>>>(zcur, delta, out, t, t_future);
  }
}